// Cluster_20753281974706
// MI455X (gfx1250) — compile-verified
//
#include <hip/hip_runtime.h>

typedef __attribute__((ext_vector_type(2))) float v2f;
typedef __attribute__((ext_vector_type(4))) float v4f;
typedef __attribute__((ext_vector_type(8))) float v8f;

#define QK_SCALE 0.17677669529663687f
#define FSQRT(x) __builtin_amdgcn_sqrtf(x)   // raw v_sqrt_f32, no IEEE fixup

// ---------------------------------------------------------------------------
// Kernel 1: one workgroup per (b,h). 1024 threads = 32 wave32's.
//  A) window-mean pool of x_k -> c_q, L2 normalize (in LDS)
//  B) aff = sc * (xf @ c_q^T) + rpb_init via V_WMMA_F32_16X16X4_F32,
//     per-row argmax -> one-hot cluster assignment (cid in LDS)
//  C) scatter-accumulate x_k / x_v / coords / counts per cluster
//     (lane-rotated ds_add_f32 -> bank-conflict/serialization free)
//  D) finalize c_q (l2norm), c_v (sum/count), centers -> workspace
// ---------------------------------------------------------------------------
__global__ __launch_bounds__(1024)
void cluster_phase1(const float* __restrict__ xk,
                    const float* __restrict__ xv,
                    const float* __restrict__ scale,
                    float* __restrict__ out,       // c_q @0, c_v @131072
                    float* __restrict__ centers)   // ws: [bh][2][64]
{
    __shared__ float cq[64 * 32];
    __shared__ float acck[64 * 32];
    __shared__ float accv[64 * 32];
    __shared__ float cnt[64], cih[64], cjw[64], invn[64], sden[64];
    __shared__ unsigned char cid[4096];

    const int tid = threadIdx.x;
    const int bh  = blockIdx.x;
    const float* __restrict__ xkb = xk + (size_t)bh * 4096 * 32;
    const float* __restrict__ xvb = xv + (size_t)bh * 4096 * 32;
    const float sc = QK_SCALE * scale[0];

    // ---- Phase A: window mean pool + zero accumulators ----
    for (int mk = tid; mk < 2048; mk += 1024) {
        const int m = mk >> 5, k = mk & 31;
        const int wi = m >> 3, wj = m & 7;
        float s = 0.f;
        const float* p = xkb + ((wi * 8) * 64 + wj * 8) * 32 + k;
        for (int a = 0; a < 8; ++a)
            for (int b = 0; b < 8; ++b)
                s += p[(a * 64 + b) * 32];
        cq[mk]   = s * (1.0f / 64.0f);
        acck[mk] = 0.f;
        accv[mk] = 0.f;
    }
    if (tid < 64) { cnt[tid] = 0.f; cih[tid] = 0.f; cjw[tid] = 0.f; }
    __syncthreads();
    if (tid < 64) {
        float ss = 0.f;
        for (int k = 0; k < 32; ++k) { float v = cq[tid * 32 + k]; ss += v * v; }
        invn[tid] = 1.0f / fmaxf(FSQRT(ss), 1e-12f);
    }
    __syncthreads();
    for (int mk = tid; mk < 2048; mk += 1024) cq[mk] *= invn[mk >> 5];
    __syncthreads();

    // ---- Phase B: WMMA affinity + argmax ----
    const int wave = tid >> 5;
    const int lane = tid & 31;
    const int hi   = lane >> 4;   // lane-half (selects K pair / row+8)
    const int lo   = lane & 15;

    for (int tt = 0; tt < 8; ++tt) {
        const int l0 = (wave * 8 + tt) * 16;          // 16-pixel tile
        // A fragments: xf[l0+lo][k], 8 K-chunks of 4 (lane-half picks K pair)
        v2f a[8];
        const int arow = l0 + lo;
        for (int c = 0; c < 8; ++c)
            a[c] = *(const v2f*)(xkb + arow * 32 + c * 4 + hi * 2);

        float best[8];
        int   bestm[8];
        for (int r = 0; r < 8; ++r) { best[r] = -3.4e38f; bestm[r] = 0; }

        for (int n0 = 0; n0 < 4; ++n0) {
            v8f acc = {0.f, 0.f, 0.f, 0.f, 0.f, 0.f, 0.f, 0.f};
            const int mcol = n0 * 16 + lo;            // cluster column
            for (int c = 0; c < 8; ++c) {
                // B = c_q^T fragment from LDS (K pair picked by lane-half)
                v2f b = *(const v2f*)(&cq[mcol * 32 + c * 4 + hi * 2]);
                acc = __builtin_amdgcn_wmma_f32_16x16x4_f32(
                          false, a[c], false, b, (short)0, acc, false, false);
            }
            // epilogue: + exp(-dist to initial window center), track argmax
            const float chi = (float)((mcol >> 3) * 8) + 3.5f;
            const float chj = (float)((mcol & 7) * 8) + 3.5f;
            for (int r = 0; r < 8; ++r) {
                const int l = l0 + r + 8 * hi;        // pixel for this D row
                const float fi = (float)(l >> 6), fj = (float)(l & 63);
                const float dh = fi - chi, dw = fj - chj;
                const float val = acc[r] * sc + __expf(-FSQRT(dh * dh + dw * dw));
                if (val > best[r]) { best[r] = val; bestm[r] = mcol; }
            }
        }
        // cross-lane argmax within each 16-lane half (columns of the tile)
        for (int r = 0; r < 8; ++r) {
            float bv = best[r];
            int   bm = bestm[r];
            for (int off = 1; off < 16; off <<= 1) {
                float ov = __shfl_xor(bv, off, 32);
                int   om = __shfl_xor(bm, off, 32);
                if (ov > bv || (ov == bv && om < bm)) { bv = ov; bm = om; }
            }
            if (lo == 0) cid[l0 + r + 8 * hi] = (unsigned char)bm;
        }
    }
    __syncthreads();

    // ---- Phase C: scatter accumulate per cluster (lane-rotated k index:
    //      lanes sharing a cluster hit 32 distinct banks -> no serialization)
    for (int l = tid; l < 4096; l += 1024) {
        const int m = cid[l];
        const float* pk = xkb + l * 32;
        const float* pv = xvb + l * 32;
        float* ak = acck + m * 32;
        float* av = accv + m * 32;
        for (int k = 0; k < 32; ++k) {
            const int kk = (k + lane) & 31;
            atomicAdd(&ak[kk], pk[kk]);
            atomicAdd(&av[kk], pv[kk]);
        }
        atomicAdd(&cnt[m], 1.0f);
        atomicAdd(&cih[m], (float)(l >> 6));
        atomicAdd(&cjw[m], (float)(l & 63));
    }
    __syncthreads();

    // ---- Phase D: finalize ----
    if (tid < 64) {
        const float c = cnt[tid];
        const float s = (c == 0.f) ? 1.f : c;
        centers[bh * 128 + tid]      = cih[tid] / s;
        centers[bh * 128 + 64 + tid] = cjw[tid] / s;
        sden[tid] = 1.0f / s;
        float ss = 0.f;
        for (int k = 0; k < 32; ++k) { float v = acck[tid * 32 + k]; ss += v * v; }
        invn[tid] = 1.0f / fmaxf(FSQRT(ss), 1e-12f);
    }
    __syncthreads();
    for (int mk = tid; mk < 2048; mk += 1024) {
        const int m = mk >> 5;
        out[bh * 2048 + mk]          = acck[mk] * invn[m];
        out[131072 + bh * 2048 + mk] = accv[mk] * sden[m];
    }
}

// ---------------------------------------------------------------------------
// Kernel 2: stream rpb = exp(-dist(pixel, center)). Pure bandwidth (64 MB).
// 1024 blocks x 256 threads; one pixel per thread, 64 clusters, b128 stores.
// ---------------------------------------------------------------------------
__global__ __launch_bounds__(256)
void cluster_phase2(const float* __restrict__ centers,
                    float* __restrict__ rpb)   // already offset to rpb base
{
    __shared__ float cx[64], cy[64];
    const int tid = threadIdx.x;
    const int bh  = blockIdx.x >> 4;
    const int l0  = (blockIdx.x & 15) * 256;
    if (tid < 64) {
        cx[tid] = centers[bh * 128 + tid];
        cy[tid] = centers[bh * 128 + 64 + tid];
    }
    __syncthreads();
    const int l = l0 + tid;
    const float fi = (float)(l >> 6), fj = (float)(l & 63);
    float* o = rpb + (size_t)bh * 262144 + (size_t)l * 64;
    for (int m = 0; m < 64; m += 4) {
        v4f v;
        for (int q = 0; q < 4; ++q) {
            const float dh = fi - cx[m + q];
            const float dw = fj - cy[m + q];
            v[q] = __expf(-FSQRT(dh * dh + dw * dw));
        }
        *(v4f*)(o + m) = v;
    }
}

extern "C" void kernel_launch(void* const* d_in, const int* in_sizes, int n_in,
                              void* d_out, int out_size, void* d_ws, size_t ws_size,
                              hipStream_t stream) {
    const float* xk    = (const float*)d_in[0];
    const float* xv    = (const float*)d_in[1];
    const float* scale = (const float*)d_in[2];
    float* out     = (float*)d_out;
    float* centers = (float*)d_ws;     // 64 * 128 floats = 32 KB

    cluster_phase1<<<64, 1024, 0, stream>>>(xk, xv, scale, out, centers);
    cluster_phase2<<<1024, 256, 0, stream>>>(centers, out + 262144);
}